// MLFF_dmirror_40810779246646
// MI455X (gfx1250) — compile-verified
//
#include <hip/hip_runtime.h>
#include <hip/hip_bf16.h>
#include <math.h>

typedef __attribute__((ext_vector_type(2))) float v2f;
typedef __attribute__((ext_vector_type(8))) float v8f;

#define BATCH   8
#define NATOMS  256
#define NNEI    100
#define NFEAT   42
#define H1DIM   64
#define H2DIM   32
#define ATOMS_TOTAL (BATCH * NATOMS)
#define TILE_M  16

// ---------------------------------------------------------------------------
// Kernel 1: per-atom MLP forward + analytic input gradient, all matmuls via
// V_WMMA_F32_16X16X4_F32 (fp32, matches reference precision).
// One wave (32 lanes) per 16-atom tile.  2048 atoms -> 128 blocks.
// ---------------------------------------------------------------------------
__global__ __launch_bounds__(32)
void mlp_fwdbwd_kernel(const float* __restrict__ image,
                       const float* __restrict__ w1, const float* __restrict__ b1,
                       const float* __restrict__ w2, const float* __restrict__ b2,
                       const float* __restrict__ w3, const float* __restrict__ b3,
                       float* __restrict__ dEi, float* __restrict__ Ei)
{
    __shared__ float Xs [TILE_M][44];      // K padded 42 -> 44 (zeros)
    __shared__ float W1s[48][H1DIM];       // rows 42..47 zero (for W1^T tiles)
    __shared__ float W2s[H1DIM][H2DIM];
    __shared__ float w3s[H2DIM];
    __shared__ float b1s[H1DIM];
    __shared__ float b2s[H2DIM];
    __shared__ float H1s[TILE_M][H1DIM];   // h1, later overwritten with G1
    __shared__ float H2s[TILE_M][H2DIM];   // h2, later overwritten with G2

    const int tid  = threadIdx.x;   // 0..31
    const int half = tid >> 4;      // 0 or 1
    const int lm   = tid & 15;
    const int base = blockIdx.x * TILE_M;

    // ---- stage weights / activations into LDS ----
    for (int i = tid; i < 48 * H1DIM; i += 32) {
        int f = i >> 6, h = i & 63;
        W1s[f][h] = (f < NFEAT) ? w1[f * H1DIM + h] : 0.0f;
    }
    for (int i = tid; i < H1DIM * H2DIM; i += 32) W2s[i >> 5][i & 31] = w2[i];
    if (tid < H2DIM) { w3s[tid] = w3[tid]; b2s[tid] = b2[tid]; }
    for (int i = tid; i < H1DIM; i += 32) b1s[i] = b1[i];
    for (int i = tid; i < TILE_M * 44; i += 32) {
        int m = i / 44, f = i - m * 44;
        Xs[m][f] = (f < NFEAT) ? image[(size_t)(base + m) * NFEAT + f] : 0.0f;
    }
    __syncthreads();

    // ---- layer 1: H1 = sigmoid(X[16x42] @ W1[42x64] + b1) ----
    for (int nt = 0; nt < 4; ++nt) {
        v8f acc = {};
        const int col = nt * 16 + lm;
        for (int k0 = 0; k0 < 44; k0 += 4) {
            v2f a, bb;
            a.x  = Xs[lm][k0 + 2 * half];
            a.y  = Xs[lm][k0 + 2 * half + 1];
            bb.x = W1s[k0 + 2 * half][col];
            bb.y = W1s[k0 + 2 * half + 1][col];
            acc = __builtin_amdgcn_wmma_f32_16x16x4_f32(false, a, false, bb,
                                                        (short)0, acc, false, false);
        }
        #pragma unroll
        for (int r = 0; r < 8; ++r) {
            int row = r + 8 * half;
            float z = acc[r] + b1s[col];
            H1s[row][col] = 1.0f / (1.0f + expf(-z));
        }
    }
    __syncthreads();

    // ---- layer 2: H2 = sigmoid(H1[16x64] @ W2[64x32] + b2) ----
    for (int nt = 0; nt < 2; ++nt) {
        v8f acc = {};
        const int col = nt * 16 + lm;
        for (int k0 = 0; k0 < 64; k0 += 4) {
            v2f a, bb;
            a.x  = H1s[lm][k0 + 2 * half];
            a.y  = H1s[lm][k0 + 2 * half + 1];
            bb.x = W2s[k0 + 2 * half][col];
            bb.y = W2s[k0 + 2 * half + 1][col];
            acc = __builtin_amdgcn_wmma_f32_16x16x4_f32(false, a, false, bb,
                                                        (short)0, acc, false, false);
        }
        #pragma unroll
        for (int r = 0; r < 8; ++r) {
            int row = r + 8 * half;
            float z = acc[r] + b2s[col];
            H2s[row][col] = 1.0f / (1.0f + expf(-z));
        }
    }
    __syncthreads();

    // ---- Ei = H2 @ w3 + b3 ----
    if (tid < TILE_M) {
        float e = b3[0];
        for (int n2 = 0; n2 < H2DIM; ++n2) e += H2s[tid][n2] * w3s[n2];
        Ei[base + tid] = e;
    }
    __syncthreads();

    // ---- G2 = w3 * h2 * (1 - h2), in place ----
    for (int i = tid; i < TILE_M * H2DIM; i += 32) {
        int m = i >> 5, n2 = i & 31;
        float h = H2s[m][n2];
        H2s[m][n2] = w3s[n2] * h * (1.0f - h);
    }
    __syncthreads();

    // ---- G1 = (G2[16x32] @ W2^T[32x64]) * h1 * (1 - h1), into H1s ----
    for (int nt = 0; nt < 4; ++nt) {
        v8f acc = {};
        const int col = nt * 16 + lm;
        for (int k0 = 0; k0 < 32; k0 += 4) {
            v2f a, bb;
            a.x  = H2s[lm][k0 + 2 * half];
            a.y  = H2s[lm][k0 + 2 * half + 1];
            bb.x = W2s[col][k0 + 2 * half];      // W2^T[k][col]
            bb.y = W2s[col][k0 + 2 * half + 1];
            acc = __builtin_amdgcn_wmma_f32_16x16x4_f32(false, a, false, bb,
                                                        (short)0, acc, false, false);
        }
        #pragma unroll
        for (int r = 0; r < 8; ++r) {
            int row = r + 8 * half;
            float h = H1s[row][col];
            H1s[row][col] = acc[r] * h * (1.0f - h);
        }
    }
    __syncthreads();

    // ---- dX = G1[16x64] @ W1^T[64x42] -> dEi ----
    for (int nt = 0; nt < 3; ++nt) {
        v8f acc = {};
        const int col = nt * 16 + lm;          // 0..47, cols 42..47 discarded
        for (int k0 = 0; k0 < 64; k0 += 4) {
            v2f a, bb;
            a.x  = H1s[lm][k0 + 2 * half];
            a.y  = H1s[lm][k0 + 2 * half + 1];
            bb.x = W1s[col][k0 + 2 * half];      // W1^T[k][col]
            bb.y = W1s[col][k0 + 2 * half + 1];
            acc = __builtin_amdgcn_wmma_f32_16x16x4_f32(false, a, false, bb,
                                                        (short)0, acc, false, false);
        }
        #pragma unroll
        for (int r = 0; r < 8; ++r) {
            int row = r + 8 * half;
            if (col < NFEAT)
                dEi[(size_t)(base + row) * NFEAT + col] = acc[r];
        }
    }
}

// ---------------------------------------------------------------------------
// Kernel 2: Etot[b] = sum over atoms of Ei  (deterministic tree reduction)
// ---------------------------------------------------------------------------
__global__ __launch_bounds__(NATOMS)
void etot_kernel(const float* __restrict__ Ei, float* __restrict__ out)
{
    __shared__ float s[NATOMS];
    const int b = blockIdx.x, t = threadIdx.x;
    s[t] = Ei[b * NATOMS + t];
    __syncthreads();
    for (int st = NATOMS / 2; st > 0; st >>= 1) {
        if (t < st) s[t] += s[t + st];
        __syncthreads();
    }
    if (t == 0) out[b] = s[0];
}

// ---------------------------------------------------------------------------
// Kernel 3: Force[b,n,d] = sum_{k,f} dEi_nb[k,f] * dfeat[b,n,k,f,d]
// One block per atom; gather dEi rows to LDS, stream dfeat coalesced.
// ---------------------------------------------------------------------------
__global__ __launch_bounds__(256)
void force_kernel(const float* __restrict__ dfeat,
                  const int*   __restrict__ neighbor,
                  const float* __restrict__ dEi,
                  float* __restrict__ out)
{
    __shared__ int   snb[NNEI];
    __shared__ float sde[NNEI * NFEAT];   // 4200 floats = 16.8 KB
    __shared__ float r0[256], r1[256], r2[256];

    const int t    = threadIdx.x;
    const int atom = blockIdx.x;          // b*NATOMS + n
    const int b    = atom / NATOMS;

    if (t < NNEI) snb[t] = neighbor[(size_t)atom * NNEI + t];
    __syncthreads();

    // gather neighbor dEi rows (index 0 == zero-pad row; j>=1 -> atom j-1)
    for (int i = t; i < NNEI * NFEAT; i += 256) {
        int k = i / NFEAT, f = i - k * NFEAT;
        int nb = snb[k];
        sde[i] = (nb == 0) ? 0.0f
                           : dEi[((size_t)b * NATOMS + (nb - 1)) * NFEAT + f];
    }
    __syncthreads();

    // stream this atom's dfeat slice: 4200 contiguous 12-byte triplets
    float a0 = 0.0f, a1 = 0.0f, a2 = 0.0f;
    const float* df = dfeat + (size_t)atom * (NNEI * NFEAT * 3);
    for (int p = t; p < NNEI * NFEAT; p += 256) {
        float c = sde[p];
        const float* v = df + (size_t)p * 3;
        a0 += c * v[0];
        a1 += c * v[1];
        a2 += c * v[2];
    }
    r0[t] = a0; r1[t] = a1; r2[t] = a2;
    __syncthreads();
    for (int st = 128; st > 0; st >>= 1) {
        if (t < st) { r0[t] += r0[t + st]; r1[t] += r1[t + st]; r2[t] += r2[t + st]; }
        __syncthreads();
    }
    if (t == 0) {
        out[BATCH + (size_t)atom * 3 + 0] = r0[0];
        out[BATCH + (size_t)atom * 3 + 1] = r1[0];
        out[BATCH + (size_t)atom * 3 + 2] = r2[0];
    }
}

// ---------------------------------------------------------------------------
extern "C" void kernel_launch(void* const* d_in, const int* in_sizes, int n_in,
                              void* d_out, int out_size, void* d_ws, size_t ws_size,
                              hipStream_t stream)
{
    const float* image    = (const float*)d_in[0];
    const float* dfeat    = (const float*)d_in[1];
    const int*   neighbor = (const int*)  d_in[2];
    // d_in[3] Egroup_weight, d_in[4] divider: unused by the reference output
    const float* w1 = (const float*)d_in[5];
    const float* b1 = (const float*)d_in[6];
    const float* w2 = (const float*)d_in[7];
    const float* b2 = (const float*)d_in[8];
    const float* w3 = (const float*)d_in[9];
    const float* b3 = (const float*)d_in[10];

    float* out    = (float*)d_out;                 // [8 Etot][8*256*3 Force]
    float* ws_dEi = (float*)d_ws;                  // 2048*42 floats
    float* ws_Ei  = ws_dEi + (size_t)ATOMS_TOTAL * NFEAT;  // 2048 floats

    mlp_fwdbwd_kernel<<<ATOMS_TOTAL / TILE_M, 32, 0, stream>>>(
        image, w1, b1, w2, b2, w3, b3, ws_dEi, ws_Ei);

    etot_kernel<<<BATCH, NATOMS, 0, stream>>>(ws_Ei, out);

    force_kernel<<<ATOMS_TOTAL, 256, 0, stream>>>(dfeat, neighbor, ws_dEi, out);
}